// spGraphKHopDecentDiffAttentionLayer_85847806313251
// MI455X (gfx1250) — compile-verified
//
#include <hip/hip_runtime.h>

#define IN_F 256
#define OUT_F 256
#define HOPS 3
#define LEAKY_ALPHA 0.2f
#define BETA 0.5f

typedef __attribute__((ext_vector_type(2))) float v2f;
typedef __attribute__((ext_vector_type(8))) float v8f;
typedef __attribute__((ext_vector_type(4))) unsigned int v4u;
typedef __attribute__((ext_vector_type(8))) int v8i;
typedef __attribute__((ext_vector_type(4))) int v4i;

// ---------------- zero scratch ----------------
__global__ void gat_zero_kernel(float* __restrict__ p, long n) {
  long i = (long)blockIdx.x * blockDim.x + threadIdx.x;
  if (i < n) p[i] = 0.0f;
}

// ---------------- H = A(N x 256) * W(256 x 256), fp32 WMMA ----------------
// Block: 256 threads = 8 waves; each block computes a 128 x 256 output tile.
// A tile lands in LDS with row stride 260 floats (260 % 64 == 4) so the
// 16-lane b64 A-fragment reads hit all 64 banks exactly once. The tile is
// DMA'd by the Tensor Data Mover (pad_enable: +4 DWORDs every 256 DWORDs
// reproduces the 260-float stride in hardware); OOB rows read as zero.
__global__ __launch_bounds__(256) void gat_gemm_wmma_kernel(
    const float* __restrict__ A, const float* __restrict__ W,
    float* __restrict__ H, int N) {
  __shared__ __align__(16) float As[128 * 260];
  const int tid = threadIdx.x;
  const int m0 = blockIdx.x * 128;

#if __has_builtin(__builtin_amdgcn_tensor_load_to_lds)
  if (tid < 32) {  // wave 0 issues the TDM descriptor; EXEC is ignored by TDM
    unsigned long long ga =
        (unsigned long long)(uintptr_t)(A + (size_t)m0 * 256);
    unsigned lds_base = (unsigned)(uintptr_t)(void*)&As[0];  // addr[31:0] = LDS offset
    unsigned rows = (unsigned)(N - m0);  // tensor_dim1; rows beyond read zero

    v4u g0;
    g0.x = 1u;                               // count=1, is_restore=0
    g0.y = lds_base;                         // lds_addr
    g0.z = (unsigned)(ga & 0xFFFFFFFFu);     // global_addr[31:0]
    g0.w = (unsigned)((ga >> 32) & 0x1FFFFFFu) | (2u << 30);  // addr[56:32], type=2

    v8i g1;
    g1[0] = (int)((2u << 16)        // data_size = 4 bytes
                  | (1u << 20)      // pad_enable
                  | (7u << 22)      // pad_interval: every 256 DWORDs
                  | (3u << 25));    // pad_amount: 4 DWORDs -> row stride 260
    g1[1] = (int)((256u & 0xFFFFu) << 16);            // tensor_dim0 = 256
    g1[2] = (int)((rows & 0xFFFFu) << 16);            // tensor_dim1 lo
    g1[3] = (int)((rows >> 16) | (256u << 16));       // tensor_dim1 hi, tile_dim0 = 256
    g1[4] = 128;                                      // tile_dim1 = 128 rows
    g1[5] = 256;                                      // tensor_dim0_stride = 256 elems
    g1[6] = 0;
    g1[7] = 0;

    v4i gz4 = {0, 0, 0, 0};
    v8i gz8 = {0, 0, 0, 0, 0, 0, 0, 0};
    __builtin_amdgcn_tensor_load_to_lds(g0, g1, gz4, gz4, gz8, 0);
    __builtin_amdgcn_s_wait_tensorcnt(0);
  }
  __syncthreads();
#else
  // Fallback: manual coalesced staging, 8192 float4, 32 per thread.
#pragma unroll 4
  for (int j = 0; j < 32; ++j) {
    int idx = j * 256 + tid;
    int r = idx >> 6;
    int c4 = idx & 63;
    int gr = m0 + r;
    if (gr >= N) gr = N - 1;
    float4 v = reinterpret_cast<const float4*>(A + (size_t)gr * 256)[c4];
    *reinterpret_cast<float4*>(&As[r * 260 + c4 * 4]) = v;
  }
  __syncthreads();
#endif

  const int wave = tid >> 5;
  const int lane = tid & 31;
  const int row16 = lane & 15;           // M (for A) / N (for B) within tile
  const int khalf = (lane >> 4) << 1;    // lanes 0-15: K+{0,1}; 16-31: K+{2,3}
  const int mloc = wave * 16;            // this wave's 16 rows of the tile
  const float* arow = &As[(mloc + row16) * 260];
  const bool interior = (m0 + 128 <= N); // block-uniform: skip store guards

  for (int nt = 0; nt < 16; ++nt) {
    v8f acc = {};
    const float* bcol = W + nt * 16 + row16;  // B[k][col] at bcol + k*256
#pragma unroll 4
    for (int k0 = 0; k0 < 256; k0 += 4) {
      v2f av = *reinterpret_cast<const v2f*>(arow + k0 + khalf);
      v2f bv;
      bv.x = bcol[(k0 + khalf) * 256];
      bv.y = bcol[(k0 + khalf + 1) * 256];
      acc = __builtin_amdgcn_wmma_f32_16x16x4_f32(
          false, av, false, bv, (short)0, acc, false, false);
    }
    // D layout: vgpr v, lane L -> row = v + 8*(L>=16), col = L&15
    const int mrow = mloc + ((lane >> 4) << 3);
    float* hout = H + (size_t)(m0 + mrow) * 256 + nt * 16 + row16;
    if (interior) {
#pragma unroll
      for (int v = 0; v < 8; ++v) hout[v * 256] = acc[v];
    } else {
#pragma unroll
      for (int v = 0; v < 8; ++v)
        if (m0 + mrow + v < N) hout[v * 256] = acc[v];
    }
  }
}

// ---------------- per-node scores: s = h . a  (one wave32 per row) --------
__global__ __launch_bounds__(256) void gat_score_kernel(
    const float* __restrict__ h, const float* __restrict__ a_src,
    const float* __restrict__ a_dst, float* __restrict__ s_src,
    float* __restrict__ s_dst, int N) {
  int gw = (int)((blockIdx.x * blockDim.x + threadIdx.x) >> 5);
  int lane = threadIdx.x & 31;
  if (gw >= N) return;
  const float4* row = reinterpret_cast<const float4*>(h + (size_t)gw * 256);
  const float4* asv = reinterpret_cast<const float4*>(a_src);
  const float4* adv = reinterpret_cast<const float4*>(a_dst);
  float ss = 0.f, sd = 0.f;
#pragma unroll
  for (int j = 0; j < 2; ++j) {
    float4 x = row[lane * 2 + j];
    float4 u = asv[lane * 2 + j];
    float4 w = adv[lane * 2 + j];
    ss += x.x * u.x + x.y * u.y + x.z * u.z + x.w * u.w;
    sd += x.x * w.x + x.y * w.y + x.z * w.z + x.w * w.w;
  }
#pragma unroll
  for (int off = 16; off > 0; off >>= 1) {
    ss += __shfl_xor(ss, off);
    sd += __shfl_xor(sd, off);
  }
  if (lane == 0) { s_src[gw] = ss; s_dst[gw] = sd; }
}

// ---------------- edge scatter: one wave32 per edge ----------------
// Gather h[dst] (1KB, coalesced), weight by exp(-leaky_relu(score)),
// scatter-add into h_prime[src] (L2-resident; 192MB L2 >> 51.2MB buffer).
__global__ __launch_bounds__(256) void gat_edge_kernel(
    const int* __restrict__ src, const int* __restrict__ dst,
    const float* __restrict__ h, const float* __restrict__ s_src,
    const float* __restrict__ s_dst, float* __restrict__ h_prime,
    float* __restrict__ e_rowsum, int E) {
  int e = (int)(blockIdx.x << 3) + (threadIdx.x >> 5);
  int lane = threadIdx.x & 31;
  if (e >= E) return;
  int s = src[e];
  int d = dst[e];
  float score = s_src[s] + s_dst[d];
  float lr = score > 0.f ? score : LEAKY_ALPHA * score;
  float w = __expf(-lr);
  if (lane == 0) atomicAdd(&e_rowsum[s], w);
  const float4* hrow = reinterpret_cast<const float4*>(h + (size_t)d * 256);
  float4 v0 = hrow[lane * 2];
  float4 v1 = hrow[lane * 2 + 1];
  float* outp = h_prime + (size_t)s * 256 + lane * 8;
  atomicAdd(outp + 0, w * v0.x);
  atomicAdd(outp + 1, w * v0.y);
  atomicAdd(outp + 2, w * v0.z);
  atomicAdd(outp + 3, w * v0.w);
  atomicAdd(outp + 4, w * v1.x);
  atomicAdd(outp + 5, w * v1.y);
  atomicAdd(outp + 6, w * v1.z);
  atomicAdd(outp + 7, w * v1.w);
}

// ---------------- combine: acc = 0.5*h'/rowsum + 0.5*acc; elu at end ------
__global__ void gat_combine_kernel(const float* __restrict__ hp,
                                   const float* __restrict__ rs,
                                   float* __restrict__ acc, long NF,
                                   int first, int last) {
  long i = (long)blockIdx.x * blockDim.x + threadIdx.x;
  if (i >= NF) return;
  float v = (1.0f - BETA) * hp[i] / rs[i >> 8];
  if (!first) v += BETA * acc[i];
  acc[i] = last ? (v > 0.f ? v : expm1f(v)) : v;
}

extern "C" void kernel_launch(void* const* d_in, const int* in_sizes, int n_in,
                              void* d_out, int out_size, void* d_ws, size_t ws_size,
                              hipStream_t stream) {
  const float* input = (const float*)d_in[0];   // (N, 256) f32
  const int* kedges = (const int*)d_in[1];      // (HOPS, 2, E) i32
  const float* W = (const float*)d_in[2];       // (768, 256) f32
  const float* a = (const float*)d_in[3];       // (1, 1536) f32

  const int N = in_sizes[0] / IN_F;
  const int E = in_sizes[1] / (2 * HOPS);
  const long NF = (long)N * OUT_F;

  float* out = (float*)d_out;
  float* ws = (float*)d_ws;
  float* h = ws;                       // N*256
  float* hp = h + NF;                  // N*256
  float* rowsum = hp + NF;             // N
  float* ssrc = rowsum + N;            // N
  float* sdst = ssrc + N;              // N

  const int zgrid_nf = (int)((NF + 255) / 256);
  const int zgrid_n = (N + 255) / 256;
  const int ggrid = (N + 127) / 128;
  const int sgrid = (N * 32 + 255) / 256;
  const int egrid = (E + 7) / 8;

  for (int i = HOPS - 1; i >= 0; --i) {
    gat_zero_kernel<<<zgrid_nf, 256, 0, stream>>>(hp, NF);
    gat_zero_kernel<<<zgrid_n, 256, 0, stream>>>(rowsum, (long)N);
    gat_gemm_wmma_kernel<<<ggrid, 256, 0, stream>>>(
        input, W + (size_t)i * IN_F * OUT_F, h, N);
    gat_score_kernel<<<sgrid, 256, 0, stream>>>(
        h, a + (size_t)(2 * i) * OUT_F, a + (size_t)(2 * i + 1) * OUT_F,
        ssrc, sdst, N);
    gat_edge_kernel<<<egrid, 256, 0, stream>>>(
        kedges + (size_t)(2 * i) * E, kedges + (size_t)(2 * i + 1) * E,
        h, ssrc, sdst, hp, rowsum, E);
    gat_combine_kernel<<<zgrid_nf, 256, 0, stream>>>(
        hp, rowsum, out, NF, i == HOPS - 1 ? 1 : 0, i == 0 ? 1 : 0);
  }
}